// GraphSAGEModel_6279242187332
// MI455X (gfx1250) — compile-verified
//
#include <hip/hip_runtime.h>

#define F 128
#define EPS_BN 1e-5f

typedef __attribute__((ext_vector_type(2))) float v2f;
typedef __attribute__((ext_vector_type(8))) float v8f;

// ---------------------------------------------------------------------------
// Degree: deg[dst] += 1 per edge (float so later divide is free)
// ---------------------------------------------------------------------------
__global__ void deg_kernel(const int* __restrict__ dst, float* __restrict__ deg, int E) {
    int e = blockIdx.x * blockDim.x + threadIdx.x;
    if (e < E) atomicAdd(&deg[dst[e]], 1.0f);
}

// ---------------------------------------------------------------------------
// Scatter-add: agg[dst, f..f+3] += x[src, f..f+3]. One thread per
// (edge, feature-quad): one b128 gather + 4 coalesced f32 atomics.
// Feature matrices (51 MB) are L2-resident, so this is atomic-throughput
// bound, not HBM bound.
// ---------------------------------------------------------------------------
__global__ void scatter_kernel(const float* __restrict__ x,
                               const int* __restrict__ src,
                               const int* __restrict__ dst,
                               float* __restrict__ agg, int E) {
    long long tid = (long long)blockIdx.x * blockDim.x + threadIdx.x;
    long long total = (long long)E * (F / 4);
    if (tid >= total) return;
    int e = (int)(tid >> 5);          // /32 quads per edge
    int q = (int)(tid & 31) << 2;     // feature quad start
    int s = src[e];
    int d = dst[e];
    float4 v = *(const float4*)&x[(long long)s * F + q];
    float* a = &agg[(long long)d * F + q];
    atomicAdd(a + 0, v.x);
    atomicAdd(a + 1, v.y);
    atomicAdd(a + 2, v.z);
    atomicAdd(a + 3, v.w);
}

// ---------------------------------------------------------------------------
// Fused SAGE linear: out = (agg/deg) @ Wl + xin @ Wr + b
// Block = 256 threads = 8 waves. Block computes rows [row0, row0+32) x all
// 128 cols; wave w computes the 32x16 output at cols [16w, 16w+16) as two
// 16x16 WMMA tiles that SHARE the B fragment (halves weight traffic) and
// give 4 independent accumulator chains for WMMA pipelining.
//
// Full-precision V_WMMA_F32_16X16X4_F32 (K-step 4, 32 steps for K=128).
// ISA 7.12.2 f32 layouts (wave32):
//   A 16x4 : lanes 0-15 -> M=lane, VGPR0=K0, VGPR1=K1 ; lanes 16-31 -> K2,K3
//   B 4x16 : lanes 0-15 -> N=lane, VGPR0=K0 row, VGPR1=K1 ; lanes16-31 K2,K3
//   C/D    : VGPR i -> M = i + 8*(lane>=16), N = lane&15
// ---------------------------------------------------------------------------
__global__ __launch_bounds__(256) void sage_gemm_kernel(
    const float* __restrict__ agg, const float* __restrict__ xin,
    const float* __restrict__ deg,
    const float* __restrict__ Wl, const float* __restrict__ Wr,
    const float* __restrict__ bias, float* __restrict__ out, int n) {

    // +4 pad: column reads stride 132 floats -> bank = (4*M + k) % 64,
    // all 32 lanes hit distinct banks. Row pitch 528 B keeps float4 stores
    // 16 B aligned.
    __shared__ __align__(16) float As[32][F + 4];
    __shared__ __align__(16) float Xs[32][F + 4];

    const int row0 = blockIdx.x * 32;
    const int tid  = threadIdx.x;

    // Stage 32x128 tiles (b128 loads, b128 LDS stores; 4 quads/thread).
    for (int i = tid; i < 32 * 32; i += 256) {
        int r = i >> 5;
        int q = (i & 31) << 2;
        int grow = row0 + r;
        float4 av = make_float4(0.f, 0.f, 0.f, 0.f);
        float4 xv = av;
        if (grow < n) {
            float inv = 1.0f / fmaxf(deg[grow], 1.0f);
            av = *(const float4*)&agg[(long long)grow * F + q];
            av.x *= inv; av.y *= inv; av.z *= inv; av.w *= inv;
            xv = *(const float4*)&xin[(long long)grow * F + q];
        }
        *(float4*)&As[r][q] = av;
        *(float4*)&Xs[r][q] = xv;
    }
    __syncthreads();

    const int wave  = tid >> 5;          // 0..7
    const int lane  = tid & 31;
    const int col0  = wave * 16;
    const int m     = lane & 15;         // M for A-frag, N for B-frag
    const int khalf = (lane >> 4) << 1;  // 0 for lanes 0-15, 2 for lanes 16-31

    v8f accA0 = {0.f, 0.f, 0.f, 0.f, 0.f, 0.f, 0.f, 0.f};
    v8f accA1 = accA0, accX0 = accA0, accX1 = accA0;

    for (int k0 = 0; k0 < F; k0 += 4) {
        int ka = k0 + khalf;             // even -> 8 B aligned LDS pair loads
        v2f aA0 = *(const v2f*)&As[m][ka];
        v2f aA1 = *(const v2f*)&As[m + 16][ka];
        v2f aX0 = *(const v2f*)&Xs[m][ka];
        v2f aX1 = *(const v2f*)&Xs[m + 16][ka];
        v2f bL, bR;
        bL.x = Wl[(long long)ka * F + col0 + m];
        bL.y = Wl[(long long)(ka + 1) * F + col0 + m];
        bR.x = Wr[(long long)ka * F + col0 + m];
        bR.y = Wr[(long long)(ka + 1) * F + col0 + m];
        accA0 = __builtin_amdgcn_wmma_f32_16x16x4_f32(
            false, aA0, false, bL, (short)0, accA0, false, false);
        accA1 = __builtin_amdgcn_wmma_f32_16x16x4_f32(
            false, aA1, false, bL, (short)0, accA1, false, false);
        accX0 = __builtin_amdgcn_wmma_f32_16x16x4_f32(
            false, aX0, false, bR, (short)0, accX0, false, false);
        accX1 = __builtin_amdgcn_wmma_f32_16x16x4_f32(
            false, aX1, false, bR, (short)0, accX1, false, false);
    }

    const int col   = col0 + m;
    const float bv  = bias[col];
    const int roff  = (lane >> 4) << 3;  // +8 for upper half-wave
    const int r0    = row0 + roff;       // tile 0 base row
    const int r1    = r0 + 16;           // tile 1 base row

    if (row0 + 32 <= n) {
        // Full tile (hot path, n tile-aligned): unconditional stores.
#pragma unroll
        for (int i = 0; i < 8; ++i) {
            out[(long long)(r0 + i) * F + col] = accA0[i] + accX0[i] + bv;
            out[(long long)(r1 + i) * F + col] = accA1[i] + accX1[i] + bv;
        }
    } else {
#pragma unroll
        for (int i = 0; i < 8; ++i) {
            if (r0 + i < n) out[(long long)(r0 + i) * F + col] = accA0[i] + accX0[i] + bv;
            if (r1 + i < n) out[(long long)(r1 + i) * F + col] = accA1[i] + accX1[i] + bv;
        }
    }
}

// ---------------------------------------------------------------------------
// BatchNorm stats: stats[f] = sum, stats[128+f] = sum of squares.
// 256 threads: thread t handles feature t&127, rows strided by 2*gridDim.
// ---------------------------------------------------------------------------
__global__ __launch_bounds__(256) void stats_kernel(const float* __restrict__ h,
                                                    float* __restrict__ stats, int n) {
    const int f    = threadIdx.x & 127;
    const int half = threadIdx.x >> 7;
    float s = 0.0f, s2 = 0.0f;
    for (int r = blockIdx.x * 2 + half; r < n; r += gridDim.x * 2) {
        float v = h[(long long)r * F + f];
        s += v;
        s2 += v * v;
    }
    __shared__ float sh[256];
    __shared__ float sh2[256];
    sh[threadIdx.x]  = s;
    sh2[threadIdx.x] = s2;
    __syncthreads();
    if (threadIdx.x < F) {
        atomicAdd(&stats[f],     sh[threadIdx.x] + sh[threadIdx.x + F]);
        atomicAdd(&stats[F + f], sh2[threadIdx.x] + sh2[threadIdx.x + F]);
    }
}

// ---------------------------------------------------------------------------
// Apply BN + ReLU in place.
// ---------------------------------------------------------------------------
__global__ void bn_relu_kernel(float* __restrict__ h, const float* __restrict__ stats,
                               const float* __restrict__ g, const float* __restrict__ be,
                               int n) {
    long long tid = (long long)blockIdx.x * blockDim.x + threadIdx.x;
    long long total = (long long)n * F;
    if (tid >= total) return;
    int f = (int)(tid & 127);
    float invn = 1.0f / (float)n;
    float mu  = stats[f] * invn;
    float var = stats[F + f] * invn - mu * mu;
    float v = h[tid];
    v = (v - mu) * rsqrtf(var + EPS_BN) * g[f] + be[f];
    h[tid] = fmaxf(v, 0.0f);
}

// ---------------------------------------------------------------------------
// Host-side orchestration (graph-capture safe: kernels + hipMemsetAsync only)
// ---------------------------------------------------------------------------
extern "C" void kernel_launch(void* const* d_in, const int* in_sizes, int n_in,
                              void* d_out, int out_size, void* d_ws, size_t ws_size,
                              hipStream_t stream) {
    const float* x   = (const float*)d_in[0];
    const float* Wl0 = (const float*)d_in[1];
    const float* Wr0 = (const float*)d_in[2];
    const float* b0  = (const float*)d_in[3];
    const float* g0  = (const float*)d_in[4];
    const float* be0 = (const float*)d_in[5];
    const float* Wl1 = (const float*)d_in[6];
    const float* Wr1 = (const float*)d_in[7];
    const float* b1  = (const float*)d_in[8];
    const float* g1  = (const float*)d_in[9];
    const float* be1 = (const float*)d_in[10];
    const float* Wl2 = (const float*)d_in[11];
    const float* Wr2 = (const float*)d_in[12];
    const float* b2  = (const float*)d_in[13];
    const int*   ei  = (const int*)d_in[14];

    const int n = in_sizes[0] / F;       // 100000
    const int E = in_sizes[14] / 2;      // 1600000
    const int* src = ei;
    const int* dst = ei + E;

    // Workspace layout
    float* agg   = (float*)d_ws;
    float* h1    = agg + (size_t)n * F;
    float* h2    = h1  + (size_t)n * F;
    float* deg   = h2  + (size_t)n * F;
    float* stats = deg + n;

    const size_t featBytes = (size_t)n * F * sizeof(float);
    const long long totEQ  = (long long)E * (F / 4);
    const int scatterGrid  = (int)((totEQ + 255) / 256);
    const int gemmGrid     = (n + 31) / 32;
    const int ewGrid       = (int)(((long long)n * F + 255) / 256);

    // Degrees (shared by all layers)
    hipMemsetAsync(deg, 0, (size_t)n * sizeof(float), stream);
    deg_kernel<<<(E + 255) / 256, 256, 0, stream>>>(dst, deg, E);

    // ---- Layer 0: x -> h1, BN+ReLU ----
    hipMemsetAsync(agg, 0, featBytes, stream);
    scatter_kernel<<<scatterGrid, 256, 0, stream>>>(x, src, dst, agg, E);
    sage_gemm_kernel<<<gemmGrid, 256, 0, stream>>>(agg, x, deg, Wl0, Wr0, b0, h1, n);
    hipMemsetAsync(stats, 0, 2 * F * sizeof(float), stream);
    stats_kernel<<<512, 256, 0, stream>>>(h1, stats, n);
    bn_relu_kernel<<<ewGrid, 256, 0, stream>>>(h1, stats, g0, be0, n);

    // ---- Layer 1: h1 -> h2, BN+ReLU ----
    hipMemsetAsync(agg, 0, featBytes, stream);
    scatter_kernel<<<scatterGrid, 256, 0, stream>>>(h1, src, dst, agg, E);
    sage_gemm_kernel<<<gemmGrid, 256, 0, stream>>>(agg, h1, deg, Wl1, Wr1, b1, h2, n);
    hipMemsetAsync(stats, 0, 2 * F * sizeof(float), stream);
    stats_kernel<<<512, 256, 0, stream>>>(h2, stats, n);
    bn_relu_kernel<<<ewGrid, 256, 0, stream>>>(h2, stats, g1, be1, n);

    // ---- Layer 2: h2 -> d_out (no BN/ReLU) ----
    hipMemsetAsync(agg, 0, featBytes, stream);
    scatter_kernel<<<scatterGrid, 256, 0, stream>>>(h2, src, dst, agg, E);
    sage_gemm_kernel<<<gemmGrid, 256, 0, stream>>>(agg, h2, deg, Wl2, Wr2, b2,
                                                   (float*)d_out, n);
}